// GraphSAGELayer_15779709845544
// MI455X (gfx1250) — compile-verified
//
#include <hip/hip_runtime.h>

// ---------------------------------------------------------------------------
// GraphSAGE layer for MI455X (gfx1250, wave32).
//  K0: zero workspace (msg_sum, deg, col stats)
//  K1: edge scatter: msg[dst] += h[src], deg[dst] += 1   (f32 atomics, L2-resident)
//  K2: out = h@Ws^T + b + (msg/deg)@Wn^T via V_WMMA_F32_16X16X4_F32 chains,
//      + per-column sum / sumsq reduction (LDS ds_add_f32 -> global atomics)
//  K3: BatchNorm (batch stats, biased var) + ReLU, in place on d_out
// ---------------------------------------------------------------------------

typedef __attribute__((ext_vector_type(2))) float v2f;
typedef __attribute__((ext_vector_type(8))) float v8f;

__global__ void sage_zero(float* __restrict__ p, long long n) {
    long long gid = (long long)blockIdx.x * blockDim.x + threadIdx.x;
    if (gid < n) p[gid] = 0.0f;
}

// One thread per (edge, feature). src/dst broadcast within the 64-thread group.
__global__ void sage_edges(const float* __restrict__ h,
                           const long long* __restrict__ src,
                           const long long* __restrict__ dst,
                           float* __restrict__ msg,
                           float* __restrict__ deg,
                           long long nedges) {
    long long gid = (long long)blockIdx.x * blockDim.x + threadIdx.x;
    long long e = gid >> 6;
    int c = (int)(gid & 63);
    if (e >= nedges) return;
    int s = (int)src[e];
    int d = (int)dst[e];
    atomicAdd(&msg[(long long)d * 64 + c], h[(long long)s * 64 + c]);
    if (c == 0) atomicAdd(&deg[d], 1.0f);
}

// 256 threads = 8 waves. Each wave owns one 16x16 output tile:
//   mtile = blockIdx.x*2 + (wave>>2), ntile = wave&3  (D_OUT=64 -> 4 n-tiles).
// Requires n_nodes % 32 == 0 (100000 = 3125*32), so EXEC is all-1s for WMMA.
__global__ __launch_bounds__(256) void sage_gemm(
    const float* __restrict__ h, const float* __restrict__ Ws,
    const float* __restrict__ bself, const float* __restrict__ Wn,
    const float* __restrict__ msg, const float* __restrict__ deg,
    float* __restrict__ out, float* __restrict__ colsum,
    float* __restrict__ colsumsq) {
    __shared__ float s_sum[64];
    __shared__ float s_sq[64];
    int tid = threadIdx.x;
    if (tid < 64) { s_sum[tid] = 0.0f; s_sq[tid] = 0.0f; }
    __syncthreads();

    int wave = tid >> 5;
    int lane = tid & 31;
    int half = lane >> 4;   // selects K pair (A) / K pair (B), and row-half of C
    int idx  = lane & 15;   // A row within tile / B,C column within tile
    int mtile = blockIdx.x * 2 + (wave >> 2);
    int ntile = wave & 3;
    int row = mtile * 16 + idx;
    int col = ntile * 16 + idx;
    int koff = half * 2;

    // --- self term: acc = h_tile(16xK) x Ws^T_tile(Kx16), K=64 in 16 steps of 4
    const float* ha  = h  + row * 64 + koff;
    const float* wsb = Ws + col * 64 + koff;   // B[k][n] = Ws[n][k]
    v8f acc = {};
#pragma unroll
    for (int k = 0; k < 16; ++k) {
        v2f a; a.x = ha[4 * k];  a.y = ha[4 * k + 1];
        v2f b; b.x = wsb[4 * k]; b.y = wsb[4 * k + 1];
        acc = __builtin_amdgcn_wmma_f32_16x16x4_f32(
            false, a, false, b, (short)0, acc, false, false);
    }

    // --- neighbor term: A = msg_row / max(deg,1), B from Wn
    float invd = 1.0f / fmaxf(deg[row], 1.0f);
    const float* ma  = msg + (long long)row * 64 + koff;
    const float* wnb = Wn  + col * 64 + koff;
#pragma unroll
    for (int k = 0; k < 16; ++k) {
        v2f a; a.x = ma[4 * k] * invd;  a.y = ma[4 * k + 1] * invd;
        v2f b; b.x = wnb[4 * k];        b.y = wnb[4 * k + 1];
        acc = __builtin_amdgcn_wmma_f32_16x16x4_f32(
            false, a, false, b, (short)0, acc, false, false);
    }

    // --- bias, store pre-BN output, accumulate column stats
    float bias = bself[col];
    float s = 0.0f, sq = 0.0f;
#pragma unroll
    for (int j = 0; j < 8; ++j) {
        float v = acc[j] + bias;
        int m = mtile * 16 + half * 8 + j;   // C/D layout: VGPR j -> row 8*half+j
        out[(long long)m * 64 + col] = v;
        s += v; sq += v * v;
    }
    atomicAdd(&s_sum[col], s);     // ds_add_f32, 4 contributors per column
    atomicAdd(&s_sq[col],  sq);
    __syncthreads();
    if (tid < 64) {
        atomicAdd(&colsum[tid],   s_sum[tid]);
        atomicAdd(&colsumsq[tid], s_sq[tid]);
    }
}

// BatchNorm (training stats, biased variance) + ReLU, in place.
__global__ void sage_bn(float* __restrict__ out,
                        const float* __restrict__ colsum,
                        const float* __restrict__ colsumsq,
                        const float* __restrict__ gamma,
                        const float* __restrict__ beta,
                        long long total, float invN) {
    long long gid = (long long)blockIdx.x * blockDim.x + threadIdx.x;
    if (gid >= total) return;
    int c = (int)(gid & 63);
    float mean = colsum[c] * invN;
    float var  = colsumsq[c] * invN - mean * mean;
    float rstd = rsqrtf(var + 1e-5f);
    float v = out[gid];
    v = (v - mean) * rstd * gamma[c] + beta[c];
    out[gid] = fmaxf(v, 0.0f);
}

extern "C" void kernel_launch(void* const* d_in, const int* in_sizes, int n_in,
                              void* d_out, int out_size, void* d_ws, size_t ws_size,
                              hipStream_t stream) {
    const float* h      = (const float*)d_in[0];
    const float* W_self = (const float*)d_in[1];
    const float* b_self = (const float*)d_in[2];
    const float* W_neigh= (const float*)d_in[3];
    const float* gamma  = (const float*)d_in[4];
    const float* beta   = (const float*)d_in[5];
    const long long* src = (const long long*)d_in[6];
    const long long* dst = (const long long*)d_in[7];

    long long n_nodes = in_sizes[0] / 64;   // 100000
    long long n_edges = in_sizes[6];        // 1600000

    // Workspace layout (floats): msg[n*64] | deg[n] | colsum[64] | colsumsq[64]
    float* msg      = (float*)d_ws;
    float* deg      = msg + n_nodes * 64;
    float* colsum   = deg + n_nodes;
    float* colsumsq = colsum + 64;
    float* out      = (float*)d_out;

    // K0: zero accumulation buffers
    long long nz = n_nodes * 64 + n_nodes + 128;
    sage_zero<<<(unsigned)((nz + 255) / 256), 256, 0, stream>>>(msg, nz);

    // K1: edge scatter (64 threads per edge)
    long long et = n_edges * 64;
    sage_edges<<<(unsigned)((et + 255) / 256), 256, 0, stream>>>(
        h, src, dst, msg, deg, n_edges);

    // K2: fused dual-GEMM + bias + column-stat reduction
    // 32 rows per block; n_nodes = 100000 is an exact multiple of 32.
    sage_gemm<<<(unsigned)(n_nodes / 32), 256, 0, stream>>>(
        h, W_self, b_self, W_neigh, msg, deg, out, colsum, colsumsq);

    // K3: BatchNorm + ReLU in place
    long long total = n_nodes * 64;
    sage_bn<<<(unsigned)((total + 255) / 256), 256, 0, stream>>>(
        out, colsum, colsumsq, gamma, beta, total, 1.0f / (float)n_nodes);
}